// SpatialVAE_66125316489697
// MI455X (gfx1250) — compile-verified
//
#include <hip/hip_runtime.h>
#include <hip/hip_bf16.h>
#include <math.h>

// ---------------------------------------------------------------------------
// SpatialVAE (graph VAE) for MI455X / gfx1250 — fp32 WMMA path.
// V_WMMA_F32_16X16X4_F32; 32x64 output tile per wave32 (8 accumulators,
// 2 A fragments share 4 B fragments -> 1.25 loads per WMMA), software-
// pipelined K loop (next-stage loads issued under current-stage WMMAs).
// Ragged N handled by pre-padded weights: the hot loop has no masks.
// ---------------------------------------------------------------------------

#define NN 50000      // nodes
#define EE 800000     // edges

typedef float v2f __attribute__((ext_vector_type(2)));
typedef float v8f __attribute__((ext_vector_type(8)));

#define WMMA_F32(a, b, c) \
    __builtin_amdgcn_wmma_f32_16x16x4_f32(false, (a), false, (b), (short)0, (c), false, false)

// ---------------- WMMA f32 GEMM: C[M x Nout] = A[MxK] * B[KxN] -------------
// N must be a multiple of 64 (B zero-padded). Nout <= N is the real output
// width. K % 4 == 0 and K >= 8. M tail (<32 rows) handled by fragment
// aliasing + store suppression (wave-uniform, EXEC stays full).
__global__ void k_gemm_wmma_f32(const float* __restrict__ A,
                                const float* __restrict__ Bm,
                                float* __restrict__ C,
                                int M, int N, int K, int Nout) {
    const int lane    = threadIdx.x & 31;
    const int wave    = threadIdx.x >> 5;
    const int groupsN = N >> 6;                    // 64-wide column groups
    const long wtile  = (long)blockIdx.x * (blockDim.x >> 5) + wave;
    const int tM      = (int)(wtile / groupsN);
    const int tN      = (int)(wtile % groupsN);
    const int r0      = tM * 32;
    if (r0 >= M) return;                           // wave-uniform branch
    const bool hasR1  = (r0 + 16 < M);
    const int r1      = hasR1 ? (r0 + 16) : r0;    // alias tail (reads in-bounds)

    const int half = lane >> 4;                    // 0: lanes 0-15, 1: 16-31
    const int l15  = lane & 15;
    const int kOff = half * 2;                     // K sub-offset 0 or 2
    const float* __restrict__ Ar0 = A + (long)(r0 + l15) * K + kOff;
    const float* __restrict__ Ar1 = A + (long)(r1 + l15) * K + kOff;
    const int col0 = tN * 64 + l15;
    const float* __restrict__ bp = Bm + (long)kOff * N + col0;
    const long bstep = (long)4 * N;

    v8f c00 = {}, c01 = {}, c02 = {}, c03 = {};    // rows r0..r0+15
    v8f c10 = {}, c11 = {}, c12 = {}, c13 = {};    // rows r1..r1+15

    // ---- pipeline prologue: stage 0 fragments ----
    v2f a0 = *(const v2f*)(Ar0);
    v2f a1 = *(const v2f*)(Ar1);
    v2f b0, b1, b2, b3;
    b0.x = bp[0];  b0.y = bp[N];
    b1.x = bp[16]; b1.y = bp[N + 16];
    b2.x = bp[32]; b2.y = bp[N + 32];
    b3.x = bp[48]; b3.y = bp[N + 48];

    for (int k = 4; k < K; k += 4) {
        // issue next-stage loads first: they retire under the 8 WMMAs below
        const v2f na0 = *(const v2f*)(Ar0 + k);
        const v2f na1 = *(const v2f*)(Ar1 + k);
        const float* __restrict__ nbp = bp + bstep;
        v2f nb0, nb1, nb2, nb3;
        nb0.x = nbp[0];  nb0.y = nbp[N];
        nb1.x = nbp[16]; nb1.y = nbp[N + 16];
        nb2.x = nbp[32]; nb2.y = nbp[N + 32];
        nb3.x = nbp[48]; nb3.y = nbp[N + 48];

        c00 = WMMA_F32(a0, b0, c00);
        c01 = WMMA_F32(a0, b1, c01);
        c02 = WMMA_F32(a0, b2, c02);
        c03 = WMMA_F32(a0, b3, c03);
        c10 = WMMA_F32(a1, b0, c10);
        c11 = WMMA_F32(a1, b1, c11);
        c12 = WMMA_F32(a1, b2, c12);
        c13 = WMMA_F32(a1, b3, c13);

        a0 = na0; a1 = na1;
        b0 = nb0; b1 = nb1; b2 = nb2; b3 = nb3;
        bp = nbp;
    }
    // ---- pipeline epilogue ----
    c00 = WMMA_F32(a0, b0, c00);
    c01 = WMMA_F32(a0, b1, c01);
    c02 = WMMA_F32(a0, b2, c02);
    c03 = WMMA_F32(a0, b3, c03);
    c10 = WMMA_F32(a1, b0, c10);
    c11 = WMMA_F32(a1, b1, c11);
    c12 = WMMA_F32(a1, b2, c12);
    c13 = WMMA_F32(a1, b3, c13);

    // C layout: VGPR i -> row base + i (lanes 0-15) / base + 8 + i (16-31)
    {
        const int rb = r0 + (half << 3);
        v8f acc[4] = {c00, c01, c02, c03};
#pragma unroll
        for (int j = 0; j < 4; ++j) {
            const int col = col0 + j * 16;
            if (col < Nout) {
#pragma unroll
                for (int i = 0; i < 8; ++i)
                    C[(long)(rb + i) * Nout + col] = acc[j][i];
            }
        }
    }
    if (hasR1) {
        const int rb = r1 + (half << 3);
        v8f acc[4] = {c10, c11, c12, c13};
#pragma unroll
        for (int j = 0; j < 4; ++j) {
            const int col = col0 + j * 16;
            if (col < Nout) {
#pragma unroll
                for (int i = 0; i < 8; ++i)
                    C[(long)(rb + i) * Nout + col] = acc[j][i];
            }
        }
    }
}

// ---------------- helpers ---------------------------------------------------
__global__ void k_zero(float* __restrict__ p, long n) {
    long i = (long)blockIdx.x * blockDim.x + threadIdx.x;
    if (i < n) p[i] = 0.0f;
}

// dst[rows x dcols] = src[rows x scols] zero-padded on the right
__global__ void k_pad(const float* __restrict__ src, float* __restrict__ dst,
                      int rows, int scols, int dcols) {
    long i = (long)blockIdx.x * blockDim.x + threadIdx.x;
    long total = (long)rows * dcols;
    if (i < total) {
        int r = (int)(i / dcols), c = (int)(i % dcols);
        dst[i] = (c < scols) ? src[(long)r * scols + c] : 0.0f;
    }
}

__global__ void k_deg(const int* __restrict__ dst, const float* __restrict__ w,
                      float* __restrict__ deg, int e) {
    int i = blockIdx.x * blockDim.x + threadIdx.x;
    if (i < e) atomicAdd(&deg[dst[i]], w[i]);
}

// deg += 1 (self loop weight), then dinv = rsqrt(deg) in place
__global__ void k_dinv(float* __restrict__ deg, int n) {
    int i = blockIdx.x * blockDim.x + threadIdx.x;
    if (i < n) {
        float d = deg[i] + 1.0f;
        deg[i] = (d > 0.0f) ? rsqrtf(d) : 0.0f;
    }
}

// agg[dst] += dinv[src]*w*dinv[dst] * h[src]   (one block per edge)
__global__ void k_agg_edges(const float* __restrict__ h, float* __restrict__ agg,
                            const int* __restrict__ src, const int* __restrict__ dst,
                            const float* __restrict__ w, const float* __restrict__ dinv,
                            int F) {
    const int e = blockIdx.x;
    const int s = src[e], d = dst[e];
    const float norm = dinv[s] * w[e] * dinv[d];
    const float* hs = h + (long)s * F;
    float* ad = agg + (long)d * F;
    for (int f = threadIdx.x; f < F; f += blockDim.x)
        atomicAdd(&ad[f], norm * hs[f]);
}

// self-loop contribution (runs after edge kernel; no atomics needed)
__global__ void k_agg_self(const float* __restrict__ h, float* __restrict__ agg,
                           const float* __restrict__ dinv, int F) {
    const int i = blockIdx.x;
    const float norm = dinv[i] * dinv[i];
    const float* hs = h + (long)i * F;
    float* ai = agg + (long)i * F;
    for (int f = threadIdx.x; f < F; f += blockDim.x)
        ai[f] += norm * hs[f];
}

__global__ void k_bias_relu(float* __restrict__ p, const float* __restrict__ b,
                            long total, int F) {
    long i = (long)blockIdx.x * blockDim.x + threadIdx.x;
    if (i < total) {
        float v = p[i] + b[(int)(i % F)];
        p[i] = v > 0.0f ? v : 0.0f;
    }
}

// mu/logvar finalize + reparameterize  (F == 64)
__global__ void k_reparam(const float* __restrict__ muagg, const float* __restrict__ lvagg,
                          const float* __restrict__ bmu, const float* __restrict__ blv,
                          const float* __restrict__ eps,
                          float* __restrict__ mu_out, float* __restrict__ lv_out,
                          float* __restrict__ Z, long total) {
    long i = (long)blockIdx.x * blockDim.x + threadIdx.x;
    if (i < total) {
        const int f = (int)(i & 63);
        const float m = muagg[i] + bmu[f];
        const float l = lvagg[i] + blv[f];
        mu_out[i] = m;
        lv_out[i] = l;
        Z[i] = m + eps[i] * expf(0.5f * l);
    }
}

// row softmax over 20 logits (TAU == 1), bias folded in
__global__ void k_softmax20(const float* __restrict__ lagg, const float* __restrict__ b,
                            float* __restrict__ B, int n) {
    int i = blockIdx.x * blockDim.x + threadIdx.x;
    if (i >= n) return;
    float v[20];
    float mx = -INFINITY;
    const float* row = lagg + (long)i * 20;
#pragma unroll
    for (int j = 0; j < 20; ++j) { v[j] = row[j] + b[j]; mx = fmaxf(mx, v[j]); }
    float s = 0.0f;
#pragma unroll
    for (int j = 0; j < 20; ++j) { v[j] = expf(v[j] - mx); s += v[j]; }
    const float inv = 1.0f / s;
    float* out = B + (long)i * 20;
#pragma unroll
    for (int j = 0; j < 20; ++j) out[j] = v[j] * inv;
}

// ---------------------------------------------------------------------------
extern "C" void kernel_launch(void* const* d_in, const int* in_sizes, int n_in,
                              void* d_out, int out_size, void* d_ws, size_t ws_size,
                              hipStream_t stream) {
    (void)in_sizes; (void)n_in; (void)out_size; (void)ws_size;

    const float* Y   = (const float*)d_in[0];   // [N,2000]
    const int*   ei  = (const int*)  d_in[1];   // [2,E]  (src row, dst row)
    const float* ew  = (const float*)d_in[2];   // [E]
    const float* Xr  = (const float*)d_in[3];   // [20,2000]
    const float* eps = (const float*)d_in[4];   // [N,64]
    const float* W1  = (const float*)d_in[5];   // [2000,512]
    const float* b1  = (const float*)d_in[6];
    const float* Wmu = (const float*)d_in[7];   // [512,64]
    const float* bmu = (const float*)d_in[8];
    const float* Wlv = (const float*)d_in[9];   // [512,64]
    const float* blv = (const float*)d_in[10];
    const float* Wd1 = (const float*)d_in[11];  // [64,512]
    const float* bd1 = (const float*)d_in[12];
    const float* Wd2 = (const float*)d_in[13];  // [512,20]
    const float* bd2 = (const float*)d_in[14];

    const int* srcI = ei;
    const int* dstI = ei + EE;

    float* out_Yhat = (float*)d_out;                 // [N,2000]
    float* out_mu   = out_Yhat + (long)NN * 2000;    // [N,64]
    float* out_lv   = out_mu   + (long)NN * 64;      // [N,64]
    float* out_B    = out_lv   + (long)NN * 64;      // [N,20]

    // ---- workspace layout (floats) ----
    float* f      = (float*)d_ws;
    float* dinv   = f;                   long o = 50048;     // deg -> dinv in place
    float* buf0   = f + o; o += (long)NN * 512;              // h_lin
    float* buf1   = f + o; o += (long)NN * 512;              // agg / H / H2
    float* buf2   = f + o; o += (long)NN * 64;               // mu_lin, later Z
    float* buf3   = f + o; o += (long)NN * 64;               // lv_lin
    float* buf4   = f + o; o += (long)NN * 64;               // mu agg
    float* buf5   = f + o; o += (long)NN * 64;               // lv agg
    float* buf6   = f + o; o += (long)NN * 20;               // logit lin
    float* buf7   = f + o; o += (long)NN * 20;               // logit agg
    float* wd2pad = f + o; o += 512 * 64;                    // Wd2 padded [512,64]
    float* xrpad  = f + o; o += 20 * 2048;                   // X_ref padded [20,2048]

    auto zero = [&](float* p, long n) {
        k_zero<<<dim3((unsigned)((n + 255) / 256)), dim3(256), 0, stream>>>(p, n);
    };
    // N must be multiple of 64 here; Nout = real output width
    auto gemm = [&](const float* A, const float* Bm, float* C,
                    int M, int Nc, int K, int Nout) {
        const int groupsN = Nc >> 6;
        const long tiles  = (long)((M + 31) >> 5) * groupsN;
        const int wpb = 4; // waves per block
        k_gemm_wmma_f32<<<dim3((unsigned)((tiles + wpb - 1) / wpb)),
                          dim3(wpb * 32), 0, stream>>>(A, Bm, C, M, Nc, K, Nout);
    };
    auto aggregate = [&](const float* h, float* agg, int F, int bt) {
        k_agg_edges<<<dim3(EE), dim3(bt), 0, stream>>>(h, agg, srcI, dstI, ew, dinv, F);
        k_agg_self <<<dim3(NN), dim3(bt), 0, stream>>>(h, agg, dinv, F);
    };

    // ---- pad ragged weight matrices (hot loops stay mask-free) ----
    k_pad<<<dim3((512 * 64 + 255) / 256), dim3(256), 0, stream>>>(Wd2, wd2pad, 512, 20, 64);
    k_pad<<<dim3((20 * 2048 + 255) / 256), dim3(256), 0, stream>>>(Xr, xrpad, 20, 2000, 2048);

    // ---- GCN normalization: deg -> dinv ----
    zero(dinv, NN);
    k_deg <<<dim3((EE + 255) / 256), dim3(256), 0, stream>>>(dstI, ew, dinv, EE);
    k_dinv<<<dim3((NN + 255) / 256), dim3(256), 0, stream>>>(dinv, NN);

    // ---- encoder layer 1: H = relu(agg(Y@W1) + b1) ----
    gemm(Y, W1, buf0, NN, 512, 2000, 512);
    zero(buf1, (long)NN * 512);
    aggregate(buf0, buf1, 512, 256);
    k_bias_relu<<<dim3((unsigned)(((long)NN * 512 + 255) / 256)), dim3(256), 0, stream>>>(
        buf1, b1, (long)NN * 512, 512);

    // ---- mu / logvar heads ----
    gemm(buf1, Wmu, buf2, NN, 64, 512, 64);
    gemm(buf1, Wlv, buf3, NN, 64, 512, 64);
    zero(buf4, (long)NN * 64);
    zero(buf5, (long)NN * 64);
    aggregate(buf2, buf4, 64, 64);
    aggregate(buf3, buf5, 64, 64);

    // ---- reparameterize: Z = mu + eps*exp(0.5*logvar) (emits mu, logvar) ----
    k_reparam<<<dim3((unsigned)(((long)NN * 64 + 255) / 256)), dim3(256), 0, stream>>>(
        buf4, buf5, bmu, blv, eps, out_mu, out_lv, buf2, (long)NN * 64);

    // ---- decoder layer 1: H2 = relu(agg(Z@Wd1) + bd1) ----
    gemm(buf2, Wd1, buf0, NN, 512, 64, 512);
    zero(buf1, (long)NN * 512);
    aggregate(buf0, buf1, 512, 256);
    k_bias_relu<<<dim3((unsigned)(((long)NN * 512 + 255) / 256)), dim3(256), 0, stream>>>(
        buf1, bd1, (long)NN * 512, 512);

    // ---- decoder layer 2 + softmax: B = softmax(agg(H2@Wd2) + bd2) ----
    gemm(buf1, wd2pad, buf6, NN, 64, 512, 20);
    zero(buf7, (long)NN * 20);
    aggregate(buf6, buf7, 20, 32);
    k_softmax20<<<dim3((NN + 255) / 256), dim3(256), 0, stream>>>(buf7, bd2, out_B, NN);

    // ---- reconstruction: Y_hat = B @ X_ref ----
    gemm(out_B, xrpad, out_Yhat, NN, 2048, 20, 2000);
}